// GAT_2963527435046
// MI455X (gfx1250) — compile-verified
//
#include <hip/hip_runtime.h>
#include <math.h>

// ---------------- problem constants (match reference) ----------------
#define NN      50000      // N_NODES
#define NE      1600000    // N_EDGES
#define IN_DIM  128
#define HID     32
#define HEADS   4
#define D1      (HEADS*HID)   // 128
#define ODIM    16
#define NEG_SLOPE 0.2f

typedef __attribute__((ext_vector_type(2))) float v2f;
typedef __attribute__((ext_vector_type(8))) float v8f;

static inline unsigned int cdiv(long long a, long long b) { return (unsigned int)((a + b - 1) / b); }

// ---------------- helpers ----------------
__device__ __forceinline__ void atomic_max_float(float* addr, float val) {
    // Sign-split trick: int ordering matches float ordering for val>=0,
    // reversed unsigned ordering for val<0. Requires init to -FLT_MAX.
    if (val >= 0.0f) atomicMax((int*)addr, __float_as_int(val));
    else             atomicMin((unsigned int*)addr, __float_as_uint(val));
}

__global__ void fill_f32(float* __restrict__ p, float v, long long n) {
    long long i = (long long)blockIdx.x * blockDim.x + threadIdx.x;
    long long stride = (long long)gridDim.x * blockDim.x;
    for (; i < n; i += stride) p[i] = v;
}

// ---------------- WMMA GEMM: Y[n_rows x n_cols] = X[n_rows x K] * W[K x n_cols] ----------
// One wave computes one 16x16 tile with V_WMMA_F32_16X16X4_F32, stepping K by 4.
// A 16x4 layout : lanes 0-15 hold (M=lane, K=0/1) in VGPR0/1; lanes 16-31 hold K=2/3.
// B 4x16 layout : lanes 0-15 hold (K=0/1, N=lane) in VGPR0/1; lanes 16-31 hold K=2/3.
// C/D 16x16     : VGPR j = row j (lanes 0-15) / row j+8 (lanes 16-31), col = lane%16.
__global__ void gemm_wmma_f32(const float* __restrict__ X, const float* __restrict__ W,
                              float* __restrict__ Y, int n_rows, int K, int n_cols) {
    const int wave = threadIdx.x >> 5;
    const int lane = threadIdx.x & 31;
    const int col_tiles = n_cols >> 4;
    const int row_tiles = n_rows >> 4;
    const int tile = blockIdx.x * (blockDim.x >> 5) + wave;
    if (tile >= row_tiles * col_tiles) return;
    const int r0 = (tile / col_tiles) << 4;
    const int c0 = (tile % col_tiles) << 4;

    const int m     = lane & 15;
    const int khalf = (lane >> 4) << 1;          // 0 for lanes 0-15, 2 for lanes 16-31

    const float* __restrict__ xrow = X + (long long)(r0 + m) * K;
    const float* __restrict__ wcol = W + c0 + m;

    v8f acc = {};
    for (int k = 0; k < K; k += 4) {
        v2f a, b;
        a.x = xrow[k + khalf];
        a.y = xrow[k + khalf + 1];
        b.x = wcol[(long long)(k + khalf) * n_cols];
        b.y = wcol[(long long)(k + khalf + 1) * n_cols];
        acc = __builtin_amdgcn_wmma_f32_16x16x4_f32(false, a, false, b,
                                                    (short)0, acc, false, false);
    }
    const int rbase = r0 + ((lane >> 4) << 3);   // +8 for upper half-wave
    const int c     = c0 + m;
#pragma unroll
    for (int j = 0; j < 8; ++j)
        Y[(long long)(rbase + j) * n_cols + c] = acc[j];
}

// ---------------- layer-1 attention coefficients: el/er [N, HEADS] ----------------
// block = 128 threads = 4 waves; wave h reduces head h of node blockIdx.x.
__global__ void attn_coef1(const float* __restrict__ feat, const float* __restrict__ al,
                           const float* __restrict__ ar, float* __restrict__ el,
                           float* __restrict__ er) {
    const int n = blockIdx.x;
    const int h = threadIdx.x >> 5;
    const int d = threadIdx.x & 31;
    const float f = feat[(long long)n * D1 + h * HID + d];
    float vl = f * al[h * HID + d];
    float vr = f * ar[h * HID + d];
#pragma unroll
    for (int off = 16; off; off >>= 1) {
        vl += __shfl_xor(vl, off, 32);
        vr += __shfl_xor(vr, off, 32);
    }
    if (d == 0) { el[n * HEADS + h] = vl; er[n * HEADS + h] = vr; }
}

// ---------------- layer-2 attention coefficients (1 head, 16 dims) ----------------
__global__ void attn_coef2(const float* __restrict__ feat, const float* __restrict__ al,
                           const float* __restrict__ ar, float* __restrict__ el,
                           float* __restrict__ er, int n_nodes) {
    const int n = blockIdx.x * blockDim.x + threadIdx.x;
    if (n >= n_nodes) return;
    float sl = 0.0f, sr = 0.0f;
#pragma unroll
    for (int d = 0; d < ODIM; ++d) {
        const float f = feat[(long long)n * ODIM + d];
        sl += f * al[d];
        sr += f * ar[d];
    }
    el[n] = sl; er[n] = sr;
}

// ---------------- edge pass 1: e = leaky_relu(el[src]+er[dst]); segment max ---------
template <int H>
__global__ void edge_attn_max(const int* __restrict__ src, const int* __restrict__ dst,
                              const float* __restrict__ el, const float* __restrict__ er,
                              float* __restrict__ e, float* __restrict__ m, long long total) {
    const long long i = (long long)blockIdx.x * blockDim.x + threadIdx.x;
    if (i >= total) return;
    const long long edge = i / H;
    const int h = (int)(i % H);
    const int sn = src[edge], dn = dst[edge];
    float v = el[sn * H + h] + er[dn * H + h];
    v = v > 0.0f ? v : NEG_SLOPE * v;
    e[i] = v;
    atomic_max_float(&m[dn * H + h], v);
}

// ---------------- edge pass 2: ex = exp(e - m[dst]); segment sum --------------------
template <int H>
__global__ void edge_attn_expsum(const int* __restrict__ dst, float* __restrict__ e,
                                 const float* __restrict__ m, float* __restrict__ s,
                                 long long total) {
    const long long i = (long long)blockIdx.x * blockDim.x + threadIdx.x;
    if (i >= total) return;
    const long long edge = i / H;
    const int h = (int)(i % H);
    const int dn = dst[edge];
    const float v = expf(e[i] - m[dn * H + h]);
    e[i] = v;                                   // overwrite e with ex
    atomicAdd(&s[dn * H + h], v);
}

// ---------------- edge aggregate, layer 1: one wave per edge, 128 dims --------------
__global__ void edge_aggregate1(const int* __restrict__ src, const int* __restrict__ dst,
                                const float* __restrict__ ex, const float* __restrict__ s,
                                const float* __restrict__ feat, float* __restrict__ out) {
    const int wave = threadIdx.x >> 5;
    const int lane = threadIdx.x & 31;
    const long long edge = (long long)blockIdx.x * (blockDim.x >> 5) + wave;
    if (edge >= NE) return;
    const int sn = src[edge], dn = dst[edge];
    const int h = lane >> 3;                    // 8 lanes (32 floats) per head
    const float alpha = ex[edge * HEADS + h] / s[dn * HEADS + h];
    const float4 f = ((const float4*)(feat + (long long)sn * D1))[lane];  // 128-bit load
    float* o = out + (long long)dn * D1 + lane * 4;
    atomicAdd(o + 0, alpha * f.x);
    atomicAdd(o + 1, alpha * f.y);
    atomicAdd(o + 2, alpha * f.z);
    atomicAdd(o + 3, alpha * f.w);
}

// ---------------- x1 = mean_h relu(rst + b1) : [N, HID] -----------------------------
__global__ void relu_mean(const float* __restrict__ rst, const float* __restrict__ b,
                          float* __restrict__ x1, long long total) {
    const long long i = (long long)blockIdx.x * blockDim.x + threadIdx.x;
    if (i >= total) return;
    const int n = (int)(i / HID), d = (int)(i % HID);
    float acc = 0.0f;
#pragma unroll
    for (int h = 0; h < HEADS; ++h) {
        const float v = rst[(long long)n * D1 + h * HID + d] + b[h * HID + d];
        acc += v > 0.0f ? v : 0.0f;
    }
    x1[i] = acc * (1.0f / HEADS);
}

// ---------------- edge aggregate, layer 2: 16 threads per edge ----------------------
__global__ void edge_aggregate2(const int* __restrict__ src, const int* __restrict__ dst,
                                const float* __restrict__ ex, const float* __restrict__ s,
                                const float* __restrict__ feat, float* __restrict__ out) {
    const long long i = (long long)blockIdx.x * blockDim.x + threadIdx.x;
    if (i >= (long long)NE * ODIM) return;
    const long long edge = i >> 4;
    const int d = (int)(i & 15);
    const int sn = src[edge], dn = dst[edge];
    const float alpha = ex[edge] / s[dn];
    atomicAdd(&out[(long long)dn * ODIM + d], alpha * feat[(long long)sn * ODIM + d]);
}

// ---------------- final: log_softmax(rst2 + b2) -------------------------------------
__global__ void logsoftmax_k(const float* __restrict__ rst, const float* __restrict__ b,
                             float* __restrict__ out, int n_nodes) {
    const int n = blockIdx.x * blockDim.x + threadIdx.x;
    if (n >= n_nodes) return;
    float x[ODIM];
    float mx = -3.402823466e38f;
#pragma unroll
    for (int d = 0; d < ODIM; ++d) {
        x[d] = rst[(long long)n * ODIM + d] + b[d];
        mx = fmaxf(mx, x[d]);
    }
    float sum = 0.0f;
#pragma unroll
    for (int d = 0; d < ODIM; ++d) sum += expf(x[d] - mx);
    const float lse = logf(sum);
#pragma unroll
    for (int d = 0; d < ODIM; ++d) out[(long long)n * ODIM + d] = x[d] - mx - lse;
}

// =====================================================================================
extern "C" void kernel_launch(void* const* d_in, const int* in_sizes, int n_in,
                              void* d_out, int out_size, void* d_ws, size_t ws_size,
                              hipStream_t stream) {
    (void)in_sizes; (void)n_in; (void)out_size; (void)ws_size;
    const float* features = (const float*)d_in[0];
    const int*   src      = (const int*)d_in[1];
    const int*   dst      = (const int*)d_in[2];
    const float* W1       = (const float*)d_in[3];
    const float* al1      = (const float*)d_in[4];
    const float* ar1      = (const float*)d_in[5];
    const float* b1       = (const float*)d_in[6];
    const float* W2       = (const float*)d_in[7];
    const float* al2      = (const float*)d_in[8];
    const float* ar2      = (const float*)d_in[9];
    const float* b2       = (const float*)d_in[10];
    float* out = (float*)d_out;

    // ---- workspace layout (floats) ----
    float* p = (float*)d_ws;
    float* feat1 = p; p += (long long)NN * D1;       // 6.4M
    float* el1   = p; p += NN * HEADS;
    float* er1   = p; p += NN * HEADS;
    float* m1    = p; p += NN * HEADS;
    float* s1    = p; p += NN * HEADS;
    float* e1    = p; p += (long long)NE * HEADS;    // 6.4M (e, then ex in-place)
    float* rst1  = p; p += (long long)NN * D1;       // 6.4M
    float* x1    = p; p += (long long)NN * HID;
    float* feat2 = p; p += (long long)NN * ODIM;
    float* el2   = p; p += NN;
    float* er2   = p; p += NN;
    float* m2    = p; p += NN;
    float* s2    = p; p += NN;
    float* e2    = p; p += (long long)NE;
    float* rst2  = p; p += (long long)NN * ODIM;

    const int B = 256;
    const float NEG_BIG = -3.402823466e38f;   // -FLT_MAX (safe for uint-min trick)

    // ---- init accumulators (ws is poisoned; must init every call) ----
    fill_f32<<<2048, B, 0, stream>>>(m1,   NEG_BIG, (long long)NN * HEADS);
    fill_f32<<<2048, B, 0, stream>>>(s1,   0.0f,    (long long)NN * HEADS);
    fill_f32<<<2048, B, 0, stream>>>(rst1, 0.0f,    (long long)NN * D1);
    fill_f32<<<2048, B, 0, stream>>>(m2,   NEG_BIG, (long long)NN);
    fill_f32<<<2048, B, 0, stream>>>(s2,   0.0f,    (long long)NN);
    fill_f32<<<2048, B, 0, stream>>>(rst2, 0.0f,    (long long)NN * ODIM);

    // ---- layer 1 ----
    {   // feat1 = features @ W1  (3125 row tiles x 8 col tiles; 8 waves/block)
        const int tiles = (NN / 16) * (D1 / 16);
        gemm_wmma_f32<<<cdiv(tiles, 8), B, 0, stream>>>(features, W1, feat1, NN, IN_DIM, D1);
    }
    attn_coef1<<<NN, 128, 0, stream>>>(feat1, al1, ar1, el1, er1);

    const long long tot1 = (long long)NE * HEADS;
    edge_attn_max<HEADS><<<cdiv(tot1, B), B, 0, stream>>>(src, dst, el1, er1, e1, m1, tot1);
    edge_attn_expsum<HEADS><<<cdiv(tot1, B), B, 0, stream>>>(dst, e1, m1, s1, tot1);
    edge_aggregate1<<<cdiv(NE, 8), B, 0, stream>>>(src, dst, e1, s1, feat1, rst1);

    relu_mean<<<cdiv((long long)NN * HID, B), B, 0, stream>>>(rst1, b1, x1, (long long)NN * HID);

    // ---- layer 2 ----
    {   // feat2 = x1 @ W2  (3125 row tiles x 1 col tile)
        const int tiles = (NN / 16) * (ODIM / 16);
        gemm_wmma_f32<<<cdiv(tiles, 8), B, 0, stream>>>(x1, W2, feat2, NN, HID, ODIM);
    }
    attn_coef2<<<cdiv(NN, B), B, 0, stream>>>(feat2, al2, ar2, el2, er2, NN);

    const long long tot2 = (long long)NE;
    edge_attn_max<1><<<cdiv(tot2, B), B, 0, stream>>>(src, dst, el2, er2, e2, m2, tot2);
    edge_attn_expsum<1><<<cdiv(tot2, B), B, 0, stream>>>(dst, e2, m2, s2, tot2);
    edge_aggregate2<<<cdiv((long long)NE * ODIM, B), B, 0, stream>>>(src, dst, e2, s2, feat2, rst2);

    logsoftmax_k<<<cdiv(NN, B), B, 0, stream>>>(rst2, b2, out, NN);
}